// RotationLoss_15951508537478
// MI455X (gfx1250) — compile-verified
//
// RotationLoss on MI455X (gfx1250).
//
// Math: loss_i depends only on t = tr(Rp^T Rt); with q = dominant eigenvector
// of Davenport's 4x4 K(M) (the SO(3)/Wahba projection, det-fix included),
// t = 4*<qp,qt>^2 - 1.  Per element: build two 4x4 symmetric K's, run a
// normalization-free power iteration on K/c + I (spectrum in [0,2], growth
// bounded, c = sqrt(3)*||M||_F), normalize once at the end, then
// chordal = 8-8d^2, cos = clip(2d^2-1), angular = acos*180/pi.
//
// Data movement: async global->LDS (ASYNCcnt) double-buffered staging; LDS
// reads at stride 9 floats are bank-conflict free (gcd(9,64)=1).
// Reduction: fixed-order trees (deterministic), two-kernel pipeline.
//
// WMMA note: each element owns a distinct 4x4 K; there is no shared-operand
// 16x16 GEMM structure, so matrix cores cannot beat wave32 VALU here.  The
// CDNA5-specific path used instead is the async global->LDS pipeline.

#include <hip/hip_runtime.h>
#include <cstdint>

#define BLOCK 256
#define CHUNK 256                    // elements per staged chunk
#define CHUNK_FLOATS (CHUNK * 9)     // 2304 floats per input array per chunk
#define MAX_GRID 1024
#define PI_ITERS 24

static_assert(BLOCK == 256, "async chunk loader assumes 256 threads");

__device__ __forceinline__ float fast_rsqrt(float x) {
#if __has_builtin(__builtin_amdgcn_rsqf)
  return __builtin_amdgcn_rsqf(x);
#else
  return rsqrtf(x);
#endif
}

__device__ __forceinline__ float fast_rcp(float x) {
#if __has_builtin(__builtin_amdgcn_rcpf)
  return __builtin_amdgcn_rcpf(x);
#else
  return 1.0f / x;
#endif
}

// Copy CHUNK_FLOATS contiguous floats from global to LDS with the CDNA5
// async path. 256 lanes x 9 B32 transfers; the instruction's offset field is
// added to BOTH the LDS and global address, so one VGPR pair serves all 9.
__device__ __forceinline__ void async_load_chunk(const float* gbase,
                                                 float* ldsbuf, uint32_t t) {
  uint32_t ldsb = (uint32_t)(uintptr_t)(ldsbuf + t);  // low 32 bits = LDS addr
  uint32_t voff = t * 4u;
#define ROT_ASYNC_LD(OFF)                                                  \
  asm volatile("global_load_async_to_lds_b32 %0, %1, %2 offset:" #OFF      \
               :: "v"(ldsb), "v"(voff), "s"(gbase) : "memory")
  ROT_ASYNC_LD(0);    ROT_ASYNC_LD(1024); ROT_ASYNC_LD(2048);
  ROT_ASYNC_LD(3072); ROT_ASYNC_LD(4096); ROT_ASYNC_LD(5120);
  ROT_ASYNC_LD(6144); ROT_ASYNC_LD(7168); ROT_ASYNC_LD(8192);
#undef ROT_ASYNC_LD
}

__device__ __forceinline__ void load_chunk(const float* __restrict__ pred,
                                           const float* __restrict__ targ,
                                           float* bp, float* bt,
                                           int64_t cb, int64_t totalFloats,
                                           uint32_t t) {
  int64_t rem = totalFloats - cb;
  if (rem >= (int64_t)CHUNK_FLOATS) {           // hot path: full chunk, async
    async_load_chunk(pred + cb, bp, t);
    async_load_chunk(targ + cb, bt, t);
  } else {                                      // cold tail: guarded copy
    uint32_t lim = (uint32_t)(rem > 0 ? rem : 0);
#pragma clang loop vectorize(disable) interleave(disable) unroll(disable)
    for (uint32_t i = t; i < lim; i += BLOCK) {
      bp[i] = pred[cb + i];
      bt[i] = targ[cb + i];
    }
  }
}

// Dominant-eigenvector quaternions of K(p), K(q): two interleaved,
// normalization-free power-iteration chains; returns <qp,qt> for unit quats.
__device__ __forceinline__ float quat_dot_so3(const float* p, const float* q) {
  float aK00 = p[0]+p[4]+p[8], aK01 = p[7]-p[5], aK02 = p[2]-p[6], aK03 = p[3]-p[1];
  float aK11 = p[0]-p[4]-p[8], aK12 = p[1]+p[3], aK13 = p[2]+p[6];
  float aK22 = p[4]-p[0]-p[8], aK23 = p[5]+p[7];
  float aK33 = p[8]-p[0]-p[4];
  float bK00 = q[0]+q[4]+q[8], bK01 = q[7]-q[5], bK02 = q[2]-q[6], bK03 = q[3]-q[1];
  float bK11 = q[0]-q[4]-q[8], bK12 = q[1]+q[3], bK13 = q[2]+q[6];
  float bK22 = q[4]-q[0]-q[8], bK23 = q[5]+q[7];
  float bK33 = q[8]-q[0]-q[4];

  float fa = 0.f, fb = 0.f;
#pragma unroll
  for (int i = 0; i < 9; ++i) { fa += p[i]*p[i]; fb += q[i]*q[i]; }
  // c >= nuclear norm (<= sqrt(3)*fro): spectrum of K/c + I lies in [0,2],
  // top eigenvalue >= 4/3  =>  bounded growth, dominant component grows.
  float ra = fast_rcp(1.7320509f * sqrtf(fa) + 1e-30f);
  float rb = fast_rcp(1.7320509f * sqrtf(fb) + 1e-30f);
  aK01 *= ra; aK02 *= ra; aK03 *= ra; aK12 *= ra; aK13 *= ra; aK23 *= ra;
  bK01 *= rb; bK02 *= rb; bK03 *= rb; bK12 *= rb; bK13 *= rb; bK23 *= rb;
  float a0 = aK00*ra + 1.f, a1 = aK11*ra + 1.f, a2 = aK22*ra + 1.f, a3 = aK33*ra + 1.f;
  float b0 = bK00*rb + 1.f, b1 = bK11*rb + 1.f, b2 = bK22*rb + 1.f, b3 = bK33*rb + 1.f;

  float aw = 1.f, ax = 0.5f, ay = 0.25f, az = 0.125f;
  float bw = 1.f, bx = 0.5f, by = 0.25f, bz = 0.125f;
#pragma unroll 4
  for (int it = 0; it < PI_ITERS; ++it) {
    float nw = a0*aw   + aK01*ax + aK02*ay + aK03*az;
    float nx = aK01*aw + a1*ax   + aK12*ay + aK13*az;
    float ny = aK02*aw + aK12*ax + a2*ay   + aK23*az;
    float nz = aK03*aw + aK13*ax + aK23*ay + a3*az;
    float mw = b0*bw   + bK01*bx + bK02*by + bK03*bz;
    float mx = bK01*bw + b1*bx   + bK12*by + bK13*bz;
    float my = bK02*bw + bK12*bx + b2*by   + bK23*bz;
    float mz = bK03*bw + bK13*bx + bK23*by + b3*bz;
    aw = nw; ax = nx; ay = ny; az = nz;
    bw = mw; bx = mx; by = my; bz = mz;
  }
  float ina = fast_rsqrt(aw*aw + ax*ax + ay*ay + az*az);
  float inb = fast_rsqrt(bw*bw + bx*bx + by*by + bz*bz);
  return (aw*bw + ax*bx + ay*by + az*bz) * ina * inb;
}

__global__ void __launch_bounds__(BLOCK)
rot_loss_partial(const float* __restrict__ pred, const float* __restrict__ targ,
                 float* __restrict__ partial, int nElem, int nChunks) {
  __shared__ float bufP[2][CHUNK_FLOATS];
  __shared__ float bufT[2][CHUNK_FLOATS];
  __shared__ float red[BLOCK];

  const uint32_t t = threadIdx.x;
  __builtin_assume(t < BLOCK);
  const uint32_t t9 = t * 9u;
  const int nb = gridDim.x;
  const int64_t totalFloats = (int64_t)nElem * 9;
  float acc = 0.f;

  int p = 0;
  int first = blockIdx.x;
  if (first < nChunks)
    load_chunk(pred, targ, bufP[0], bufT[0],
               (int64_t)first * CHUNK_FLOATS, totalFloats, t);

  for (int c = first; c < nChunks; c += nb) {
    // Own async loads done, then collective: buf[p] fully populated; also
    // guarantees nobody still reads buf[p^1] from the previous round.
    asm volatile("s_wait_asynccnt 0" ::: "memory");
    __syncthreads();

    int cn = c + nb;                       // prefetch next chunk -> buf[p^1]
    if (cn < nChunks)
      load_chunk(pred, targ, bufP[p ^ 1], bufT[p ^ 1],
                 (int64_t)cn * CHUNK_FLOATS, totalFloats, t);

    int e = c * CHUNK + (int)t;
    if (e < nElem) {
      float mp[9], mt[9];
      const float* bp = bufP[p];
      const float* bt = bufT[p];
#pragma unroll
      for (int k = 0; k < 9; ++k) {        // stride-9: bank-conflict free
        mp[k] = bp[t9 + k];
        mt[k] = bt[t9 + k];
      }
      float d  = quat_dot_so3(mp, mt);
      float d2 = d * d;
      float chordal = 8.f - 8.f * d2;            // 6 - 2*tr, tr = 4d^2-1
      float ct = 2.f * d2 - 1.f;                 // (tr-1)/2
      ct = fminf(fmaxf(ct, -1.f + 1e-7f), 1.f - 1e-7f);
      float ang = acosf(ct) * 57.29577951308232f;
      acc += chordal + 0.1f * ang;
    }
    p ^= 1;
  }

  // deterministic block reduction
  red[t] = acc;
  __syncthreads();
  for (int s = BLOCK / 2; s > 0; s >>= 1) {
    if (t < (uint32_t)s) red[t] += red[t + s];
    __syncthreads();
  }
  if (t == 0) partial[blockIdx.x] = red[0];
}

__global__ void __launch_bounds__(BLOCK)
rot_loss_finalize(const float* __restrict__ partial, float* __restrict__ out,
                  int n, float invB) {
  __shared__ float red[BLOCK];
  float s = 0.f;
  for (int i = threadIdx.x; i < n; i += BLOCK) s += partial[i];
  red[threadIdx.x] = s;
  __syncthreads();
  for (int k = BLOCK / 2; k > 0; k >>= 1) {
    if ((int)threadIdx.x < k) red[threadIdx.x] += red[threadIdx.x + k];
    __syncthreads();
  }
  if (threadIdx.x == 0) out[0] = red[0] * invB;
}

extern "C" void kernel_launch(void* const* d_in, const int* in_sizes, int n_in,
                              void* d_out, int out_size, void* d_ws, size_t ws_size,
                              hipStream_t stream) {
  const float* pred = (const float*)d_in[0];
  const float* targ = (const float*)d_in[1];
  float* out      = (float*)d_out;
  float* partial  = (float*)d_ws;

  int nElem   = in_sizes[0] / 9;
  int nChunks = (nElem + CHUNK - 1) / CHUNK;
  int grid    = nChunks < MAX_GRID ? nChunks : MAX_GRID;
  if (grid < 1) grid = 1;

  rot_loss_partial<<<grid, BLOCK, 0, stream>>>(pred, targ, partial, nElem, nChunks);
  rot_loss_finalize<<<1, BLOCK, 0, stream>>>(partial, out, grid,
                                             1.0f / (float)nElem);
}